// GCNModel_65678639891196
// MI455X (gfx1250) — compile-verified
//
#include <hip/hip_runtime.h>
#include <hip/hip_bf16.h>

typedef __attribute__((ext_vector_type(2))) float v2f;
typedef __attribute__((ext_vector_type(8))) float v8f;

#define N_NODES 50000
#define N_EDGES 800000
#define F_IN    128
#define F_HID   128
#define F_OUT   32

// ---------------------------------------------------------------------------
// Zero-fill workspace regions (graph-capture safe; no hipMemset).
__global__ void zero_kernel(float* __restrict__ p, int n) {
    int t = blockIdx.x * blockDim.x + threadIdx.x;
    if (t < n) p[t] = 0.0f;
}

// ---------------------------------------------------------------------------
// Degree accumulation: out_deg[src]++ , in_deg[dst]++
__global__ void degree_kernel(const int* __restrict__ src, const int* __restrict__ dst,
                              float* __restrict__ outdeg, float* __restrict__ indeg, int nE) {
    int t = blockIdx.x * blockDim.x + threadIdx.x;
    if (t < nE) {
        atomicAdd(&outdeg[src[t]], 1.0f);
        atomicAdd(&indeg [dst[t]], 1.0f);
    }
}

// d = deg > 0 ? deg^-1/2 : 1.0   (in place; outdeg & indeg are contiguous)
__global__ void invsqrt_kernel(float* __restrict__ d, int n) {
    int t = blockIdx.x * blockDim.x + threadIdx.x;
    if (t < n) {
        float v = d[t];
        d[t] = (v > 0.0f) ? rsqrtf(v) : 1.0f;
    }
}

// ---------------------------------------------------------------------------
// out[r, f] = (sum_k in[r,k] * W[k,f]) * dscale[r]
// Exact f32 math on the matrix pipe: V_WMMA_F32_16X16X4_F32, K stepped by 4.
// FOUT is compile-time so every W access is base + immediate offset.
// One block = one 16-row tile; wave w owns columns [16w, 16w+16).
// blockDim.x = 32 * (FOUT/16).
template <int FOUT>
__global__ void gemm_wmma_scale(const float* __restrict__ in, const float* __restrict__ W,
                                const float* __restrict__ dscale, float* __restrict__ out) {
    __shared__ float tileA[16 * 132];          // stride 132 -> conflict-free column reads
    const int row0 = blockIdx.x * 16;

    // cooperative, coalesced stage of the 16x128 input tile (32-bit indexing)
    const float* inBase = in + row0 * F_IN;
    for (int idx = threadIdx.x; idx < 16 * F_IN; idx += blockDim.x) {
        int r = idx >> 7, c = idx & 127;
        tileA[r * 132 + c] = inBase[r * F_IN + c];
    }
    __syncthreads();

    const int lane = threadIdx.x & 31;
    const int wv   = threadIdx.x >> 5;
    const int f0   = wv * 16;
    const int half = lane >> 4;                // 0: K-pair {0,1}; 1: K-pair {2,3}
    const int ml   = lane & 15;                // M for A / N for B
    const float* arow = &tileA[ml * 132 + half * 2];
    const float* wcol = W + half * 2 * FOUT + f0 + ml;   // per-lane base; rest immediates

    v8f acc = {};
    #pragma unroll
    for (int k = 0; k < F_IN; k += 4) {
        v2f a; a[0] = arow[k];
               a[1] = arow[k + 1];
        v2f b; b[0] = wcol[k * FOUT];
               b[1] = wcol[k * FOUT + FOUT];
        acc = __builtin_amdgcn_wmma_f32_16x16x4_f32(
                  /*neg_a=*/false, a, /*neg_b=*/false, b,
                  /*c_mod=*/(short)0, acc, /*reuse_a=*/false, /*reuse_b=*/false);
    }

    float* outBase = out + (row0 + half * 8) * FOUT + f0 + ml;
    const float* dsBase = dscale + row0 + half * 8;
    #pragma unroll
    for (int i = 0; i < 8; ++i) {              // VGPR i holds rows i and 8+i
        outBase[i * FOUT] = acc[i] * dsBase[i];
    }
}

// ---------------------------------------------------------------------------
// agg[dst[e], :] += h[src[e], :]   — float4 gather + 4 scalar f32 atomics.
// For F=128 a wave reads 512 contiguous bytes of one source row.
template <int F>
__global__ void spmm_scatter(const float* __restrict__ h, const int* __restrict__ src,
                             const int* __restrict__ dst, float* __restrict__ agg, int nE) {
    constexpr int QP = F / 4;                  // float4 chunks per edge (power of two)
    int t = blockIdx.x * blockDim.x + threadIdx.x;
    if (t >= nE * QP / QP * QP ? t >= nE * QP : false) return; // (folded below)
    if (t >= nE * QP) return;
    int e = t / QP;
    int q = t - e * QP;
    int s = src[e], d = dst[e];
    const float4 v = *(const float4*)(h + s * F + q * 4);
    float* p = agg + d * F + q * 4;
    atomicAdd(p + 0, v.x);
    atomicAdd(p + 1, v.y);
    atomicAdd(p + 2, v.z);
    atomicAdd(p + 3, v.w);
}

// h1 = relu(agg * d_in[node] + b[f])          (n = N_NODES*128 elements)
__global__ void bias_relu_kernel(const float* __restrict__ agg, const float* __restrict__ din,
                                 const float* __restrict__ b, float* __restrict__ out, int n) {
    int t = blockIdx.x * blockDim.x + threadIdx.x;
    if (t < n) {
        int node = t >> 7, f = t & 127;
        float v = agg[t] * din[node] + b[f];
        out[t] = v > 0.0f ? v : 0.0f;
    }
}

// out = agg * d_in[node] + b[c]               (n = N_NODES*32 elements)
__global__ void bias_kernel(const float* __restrict__ agg, const float* __restrict__ din,
                            const float* __restrict__ b, float* __restrict__ out, int n) {
    int t = blockIdx.x * blockDim.x + threadIdx.x;
    if (t < n) {
        int node = t >> 5, c = t & 31;
        out[t] = agg[t] * din[node] + b[c];
    }
}

// ---------------------------------------------------------------------------
extern "C" void kernel_launch(void* const* d_in, const int* in_sizes, int n_in,
                              void* d_out, int out_size, void* d_ws, size_t ws_size,
                              hipStream_t stream) {
    (void)in_sizes; (void)n_in; (void)out_size; (void)ws_size;

    const float* X  = (const float*)d_in[0];   // [50000,128]
    // d_in[1] = edge_encodings: unused by the reference
    const float* W1 = (const float*)d_in[2];   // [128,128]
    const float* b1 = (const float*)d_in[3];   // [128]
    const float* W2 = (const float*)d_in[4];   // [128,32]
    const float* b2 = (const float*)d_in[5];   // [32]
    const int*   ei = (const int*)  d_in[6];   // [2,800000]
    const int* src = ei;
    const int* dst = ei + N_EDGES;

    // workspace layout (floats): [outdeg N][indeg N][bufA N*128][bufB N*128]
    float* ws     = (float*)d_ws;
    float* outdeg = ws;                        // -> d_out norm after invsqrt
    float* indeg  = ws + N_NODES;              // -> d_in  norm after invsqrt
    float* bufA   = ws + 2 * N_NODES;          // h0, then h1 (reuse)
    float* bufB   = bufA + N_NODES * F_HID;    // agg1, then g + agg2 (reuse)
    float* gbuf   = bufB;                      // [N,32] layer-2 pre-agg
    float* agg2   = bufB + N_NODES * F_OUT;

    const int TB = 256;

    // 1) degrees + normalizers
    zero_kernel<<<(2 * N_NODES + TB - 1) / TB, TB, 0, stream>>>(outdeg, 2 * N_NODES);
    degree_kernel<<<(N_EDGES + TB - 1) / TB, TB, 0, stream>>>(src, dst, outdeg, indeg, N_EDGES);
    invsqrt_kernel<<<(2 * N_NODES + TB - 1) / TB, TB, 0, stream>>>(ws, 2 * N_NODES);

    // 2) layer 1: h0 = (X @ W1) * d_out    (WMMA f32, 8 waves/block, 16 cols/wave)
    gemm_wmma_scale<F_HID><<<N_NODES / 16, 32 * (F_HID / 16), 0, stream>>>(X, W1, outdeg, bufA);

    // 3) SpMM1: agg1[dst] += h0[src]
    zero_kernel<<<(N_NODES * F_HID + TB - 1) / TB, TB, 0, stream>>>(bufB, N_NODES * F_HID);
    {
        int total = N_EDGES * (F_HID / 4);
        spmm_scatter<F_HID><<<(total + TB - 1) / TB, TB, 0, stream>>>(bufA, src, dst, bufB, N_EDGES);
    }

    // 4) h1 = relu(agg1 * d_in + b1)  (bufB -> bufA, frees bufB)
    bias_relu_kernel<<<(N_NODES * F_HID + TB - 1) / TB, TB, 0, stream>>>(
        bufB, indeg, b1, bufA, N_NODES * F_HID);

    // 5) layer 2: g = (h1 @ W2) * d_out    (2 waves/block)
    gemm_wmma_scale<F_OUT><<<N_NODES / 16, 32 * (F_OUT / 16), 0, stream>>>(bufA, W2, outdeg, gbuf);

    // 6) SpMM2: agg2[dst] += g[src]
    zero_kernel<<<(N_NODES * F_OUT + TB - 1) / TB, TB, 0, stream>>>(agg2, N_NODES * F_OUT);
    {
        int total = N_EDGES * (F_OUT / 4);
        spmm_scatter<F_OUT><<<(total + TB - 1) / TB, TB, 0, stream>>>(gbuf, src, dst, agg2, N_EDGES);
    }

    // 7) out = agg2 * d_in + b2
    bias_kernel<<<(N_NODES * F_OUT + TB - 1) / TB, TB, 0, stream>>>(
        agg2, indeg, b2, (float*)d_out, N_NODES * F_OUT);
}